// DownsampleWrappedTranslateAugment_4423816314957
// MI455X (gfx1250) — compile-verified
//
#include <hip/hip_runtime.h>
#include <hip/hip_bf16.h>

// DownsampleWrappedTranslateAugment for MI455X (gfx1250).
// B=128, C=3, H=W=224 -> 4x4 area pool -> 56x56, then per-batch cyclic shift:
//   out[b,c,(p+dy)%56,(q+dx)%56] = mean4x4(x[b,c,4p:4p+4,4q:4q+4])
//   dy = W_shifts[b] (rows), dx = H_shifts[b] (cols)   [reference swaps them]
//
// Pure streaming: 77 MB read-once + 4.8 MB write-once => ~3.5 us floor at
// 23.3 TB/s; compute (~19 MFLOP) is free, so WMMA is inapplicable. We use the
// CDNA5 async data-movement path: global_load_async_to_lds_b128 (ASYNCcnt)
// stages each contiguous 32-row tile into LDS; since the instruction's
// IOFFSET is added to BOTH the LDS and global addresses, all 7 chunk copies
// share one (lds, vaddr) register pair and differ only by offset: immediates.
// Non-temporal hints keep the one-touch stream from rinsing L2.

#define BDIM 256
#define IN_W 224
#define OUT_W 56
#define TILE_IN_ROWS 32                       // 8 pooled rows per tile
#define TILE_FLOATS (TILE_IN_ROWS * IN_W)     // 7168 floats = 28,672 bytes
#define TILES_PER_PLANE 7                     // 224 / 32
#define CHUNK_BYTES (BDIM * 16)               // 4096 B staged per issue

__device__ __forceinline__ unsigned lds_byte_addr(void* p) {
  // generic -> LDS address space, then truncate to the 32-bit LDS byte address
  return (unsigned)(size_t)(__attribute__((address_space(3))) void*)p;
}

__global__ __launch_bounds__(BDIM) void
DownsampleWrappedTranslate_kernel(const float* __restrict__ images,
                                  const int* __restrict__ H_shifts,
                                  const int* __restrict__ W_shifts,
                                  float* __restrict__ out) {
  __shared__ float lds[TILE_FLOATS];

  const int tid   = threadIdx.x;
  const int tile  = blockIdx.x % TILES_PER_PLANE;
  const int plane = blockIdx.x / TILES_PER_PLANE;   // b*3 + c
  const int b     = plane / 3;

  // Per-lane base addresses for the staging copy (16 B per lane per issue).
  const float* gbase = images + (size_t)plane * (IN_W * IN_W) +
                       (size_t)tile * TILE_FLOATS + tid * 4;
  const unsigned lbase = lds_byte_addr(&lds[tid * 4]);

  // ---- Async stage: 7 x 4096 B chunks; IOFFSET advances LDS & global ----
#pragma unroll
  for (int i = 0; i < TILE_FLOATS / (BDIM * 4); ++i) {
    asm volatile(
        "global_load_async_to_lds_b128 %0, %1, off offset:%2 th:TH_LOAD_NT"
        :
        : "v"(lbase), "v"(gbase), "i"(i * CHUNK_BYTES)
        : "memory");
  }
  // Wait for this wave's async copies, then make all waves' data visible.
  asm volatile("s_wait_asynccnt 0x0" ::: "memory");
  __syncthreads();

  // ---- 4x4 reduce from LDS + rotated non-temporal scatter ----
  const int dy = W_shifts[b];
  const int dx = H_shifts[b];
  float* obase = out + (size_t)plane * (OUT_W * OUT_W);
  const int prow0 = tile * (TILE_IN_ROWS / 4);      // first pooled row of tile

#pragma unroll 2
  for (int o = tid; o < (TILE_IN_ROWS / 4) * OUT_W; o += BDIM) {
    const int pl = o / OUT_W;                       // local pooled row 0..7
    const int q  = o - pl * OUT_W;                  // pooled col 0..55
    const float* src = &lds[(pl * 4) * IN_W + q * 4];
    const float4 r0 = *(const float4*)(src);
    const float4 r1 = *(const float4*)(src + IN_W);
    const float4 r2 = *(const float4*)(src + 2 * IN_W);
    const float4 r3 = *(const float4*)(src + 3 * IN_W);
    float s = ((r0.x + r0.y) + (r0.z + r0.w)) +
              ((r1.x + r1.y) + (r1.z + r1.w)) +
              ((r2.x + r2.y) + (r2.z + r2.w)) +
              ((r3.x + r3.y) + (r3.z + r3.w));
    s *= 0.0625f;

    int a = prow0 + pl + dy; if (a >= OUT_W) a -= OUT_W;   // p,dy < 56
    int j = q + dx;          if (j >= OUT_W) j -= OUT_W;   // q,dx < 56
    __builtin_nontemporal_store(s, &obase[a * OUT_W + j]);
  }
}

extern "C" void kernel_launch(void* const* d_in, const int* in_sizes, int n_in,
                              void* d_out, int out_size, void* d_ws,
                              size_t ws_size, hipStream_t stream) {
  const float* images   = (const float*)d_in[0];
  const int*   H_shifts = (const int*)d_in[1];
  const int*   W_shifts = (const int*)d_in[2];
  float*       out      = (float*)d_out;

  const int B = 128, C = 3;
  dim3 grid(B * C * TILES_PER_PLANE);   // 2688 blocks
  dim3 block(BDIM);
  hipLaunchKernelGGL(DownsampleWrappedTranslate_kernel, grid, block, 0, stream,
                     images, H_shifts, W_shifts, out);
}